// nonparametric_classifier_67886253080994
// MI455X (gfx1250) — compile-verified
//
#include <hip/hip_runtime.h>
#include <hip/hip_bf16.h>
#include <math.h>
#include <stdint.h>

// Problem sizes (fixed by the reference)
#define BB 128
#define DD 1024
#define CC 32768
#define INV_TAU 20.0f      // 1/0.05
#define THRESH  20.0f      // LAMBDA0/TAU

#define KC 64              // K-chunk staged in LDS
#define LSTR 68            // LDS row stride (floats): 68 mod 64 = 4 -> conflict-free b64 reads

typedef __attribute__((ext_vector_type(2))) float v2f;
typedef __attribute__((ext_vector_type(8))) float v8f;

// ---------------------------------------------------------------------------
// ws layout (floats): [0,128) rowsumexp  [128,256) cnt_sel  [256,384) sum_sel
//                     [384,512) loss_row
// ---------------------------------------------------------------------------

__global__ __launch_bounds__(512) void init_ws_kernel(float* ws) {
    ws[threadIdx.x] = 0.0f;
}

// 8 waves per block; wave w computes scores for all 128 rows x 16 columns
// (cols blockIdx.x*128 + w*16 ...) via V_WMMA_F32_16X16X4_F32.
// inputs (A) is staged through LDS with GLOBAL_LOAD_ASYNC_TO_LDS_B128 so each
// block reads it exactly once (128MB total L2 traffic); em (B) is streamed
// from HBM exactly once (128MB).
__global__ __launch_bounds__(256) void gemm_stats_kernel(
    const float* __restrict__ inputs,   // [B, D]
    const float* __restrict__ em,       // [C, D]
    float* __restrict__ rowsumexp,      // [B]
    float* __restrict__ cntsel,        // [B]
    float* __restrict__ sumsel)        // [B]
{
    __shared__ float sA[BB * LSTR];     // 128 x 68 floats = 34816 B

    const int tid  = threadIdx.x;
    const int wv   = tid >> 5;
    const int lane = tid & 31;
    const int n    = lane & 15;          // column within tile / row M for A-frag
    const int koff = (lane >> 4) << 1;   // K-pair select: 0 or 2

    const int col = blockIdx.x * 128 + wv * 16 + n;
    const float* __restrict__ bbase = em + (size_t)col * DD + koff;

    v8f acc[8];
#pragma unroll
    for (int rt = 0; rt < 8; ++rt) acc[rt] = (v8f){};

    const int srow  = tid >> 1;          // staging: 2 threads per row
    const int sbase = (tid & 1) * (KC / 2);
    // LDS byte address = low 32 bits of the generic pointer (aperture rule)
    const uint32_t lbase = (uint32_t)(size_t)(&sA[srow * LSTR + sbase]);

    for (int kc = 0; kc < DD; kc += KC) {
        __syncthreads();                 // previous chunk fully consumed
        // stage inputs[:, kc:kc+KC] -> LDS, async (no VGPR round-trip):
        // each thread issues 8 x GLOBAL_LOAD_ASYNC_TO_LDS_B128
        {
            uint64_t gbase = (uint64_t)(size_t)(inputs + (size_t)srow * DD + kc + sbase);
#pragma unroll
            for (int i = 0; i < (KC / 2) / 4; ++i) {
                uint32_t l = lbase + 16u * i;
                uint64_t g = gbase + 16u * i;
                asm volatile("global_load_async_to_lds_b128 %0, %1, off"
                             :: "v"(l), "v"(g) : "memory");
            }
            asm volatile("s_wait_asynccnt 0x0" ::: "memory");
        }
        __syncthreads();

        for (int k0 = 0; k0 < KC; k0 += 4) {
            v2f bfrag = *(const v2f*)(bbase + kc + k0);
            v2f afrag[8];
#pragma unroll
            for (int rt = 0; rt < 8; ++rt)
                afrag[rt] = *(const v2f*)(&sA[(rt * 16 + n) * LSTR + k0 + koff]);
#pragma unroll
            for (int rt = 0; rt < 8; ++rt) {
                // D = A(16x4) x B(4x16) + C  — fp32 WMMA
                acc[rt] = __builtin_amdgcn_wmma_f32_16x16x4_f32(
                    false, afrag[rt], false, bfrag, (short)0, acc[rt], false, false);
            }
        }
    }

    // D layout: VGPR r -> row (rt*16 + r + 8*(lane>=16)); lane%16 -> column.
    const int rowAdd = (lane >= 16) ? 8 : 0;
#pragma unroll
    for (int rt = 0; rt < 8; ++rt) {
#pragma unroll
        for (int r = 0; r < 8; ++r) {
            float s   = acc[rt][r] * INV_TAU;
            float e   = __expf(s - 20.0f);           // scores <= 20 (unit norms)
            float sel = (s > THRESH) ? 1.0f : 0.0f;
            float ssm = sel * s;
            // reduce across the 16 lanes of this half-wave (columns)
#pragma unroll
            for (int m = 8; m >= 1; m >>= 1) {
                e   += __shfl_xor(e,   m, 32);
                sel += __shfl_xor(sel, m, 32);
                ssm += __shfl_xor(ssm, m, 32);
            }
            if (n == 0) {
                int row = rt * 16 + r + rowAdd;
                atomicAdd(&rowsumexp[row], e);
                atomicAdd(&cntsel[row],   sel);
                atomicAdd(&sumsel[row],   ssm);
            }
        }
    }
}

// One wave per sample row: s_target dot product + closed-form loss row.
__global__ __launch_bounds__(32) void finalize_kernel(
    const float* __restrict__ inputs,
    const float* __restrict__ em,
    const int*   __restrict__ targets,
    const float* __restrict__ rowsumexp,
    const float* __restrict__ cntsel,
    const float* __restrict__ sumsel,
    float* __restrict__ out,            // d_out: [0]=loss, [1..128]=ks
    float* __restrict__ lossrow)
{
    const int b    = blockIdx.x;
    const int lane = threadIdx.x;
    const int y    = targets[b];
    const float* x = inputs + (size_t)b * DD;
    const float* w = em     + (size_t)y * DD;

    float dot = 0.0f;
    for (int d = lane; d < DD; d += 32) dot = fmaf(x[d], w[d], dot);
#pragma unroll
    for (int m = 16; m >= 1; m >>= 1) dot += __shfl_xor(dot, m, 32);

    if (lane == 0) {
        float s_t    = dot * INV_TAU;
        float cnt    = cntsel[b];
        float sumexp = rowsumexp[b];
        float ssel   = sumsel[b];

        float lse    = 20.0f + logf(sumexp);        // logsumexp with fixed shift
        float logp_t = s_t - lse;

        // adaptive k factor:  k = ks*mask + (1-mask)*2;  k = 1/(k ln k);  k *= mask
        float mask = (cnt > 1.0f) ? 1.0f : 0.0f;
        float kk   = mask * cnt + (1.0f - mask) * 2.0f;
        float kfac = mask / (kk * logf(kk));

        // sum over selected cols of logp, excluding target if it was selected
        float S_logp   = ssel - cnt * lse;
        float sel_excl = S_logp - ((s_t > THRESH) ? logp_t : 0.0f);

        lossrow[b]  = -(kfac * sel_excl + logp_t);
        out[1 + b]  = cnt;                          // ks output
    }
}

__global__ __launch_bounds__(128) void loss_reduce_kernel(
    const float* __restrict__ lossrow, float* __restrict__ out)
{
    __shared__ float sm[128];
    int t = threadIdx.x;
    sm[t] = lossrow[t];
    __syncthreads();
    for (int s = 64; s > 0; s >>= 1) {
        if (t < s) sm[t] += sm[t + s];
        __syncthreads();
    }
    if (t == 0) out[0] = sm[0] * (1.0f / (float)BB);
}

// Copy em -> new_em.  d_out+129 floats is only 4B-aligned, so: wide loads,
// scalar stores.
__global__ __launch_bounds__(256) void copy_em_kernel(
    const float4* __restrict__ src, float* __restrict__ dst, int n4)
{
    int i      = blockIdx.x * blockDim.x + threadIdx.x;
    int stride = gridDim.x * blockDim.x;
    for (; i < n4; i += stride) {
        float4 v = src[i];
        size_t o = (size_t)i * 4;
        dst[o + 0] = v.x; dst[o + 1] = v.y; dst[o + 2] = v.z; dst[o + 3] = v.w;
    }
}

// Sequential EMA scatter with renorm. Single block of 1024 threads; thread t
// owns element t of every row, so duplicate targets chain correctly with no
// cross-thread global hazards; only the norm reduction needs __syncthreads.
__global__ __launch_bounds__(1024) void ema_update_kernel(
    const float* __restrict__ inputs,
    const int*   __restrict__ targets,
    const int*   __restrict__ epoch_p,
    float* __restrict__ out_em)
{
    __shared__ float red[32];
    const int t = threadIdx.x;
    const float mu  = fminf(0.4f / 60.0f * (float)(epoch_p[0] + 1), 1.0f);
    const float omu = 1.0f - mu;

    for (int b = 0; b < BB; ++b) {
        const int y = targets[b];
        float v = mu * out_em[(size_t)y * DD + t] + omu * inputs[(size_t)b * DD + t];

        float sq = v * v;
#pragma unroll
        for (int m = 16; m >= 1; m >>= 1) sq += __shfl_xor(sq, m, 32);
        if ((t & 31) == 0) red[t >> 5] = sq;
        __syncthreads();
        if (t < 32) {
            float p = red[t];
#pragma unroll
            for (int m = 16; m >= 1; m >>= 1) p += __shfl_xor(p, m, 32);
            red[t] = p;
        }
        __syncthreads();
        float inv = rsqrtf(red[0]);
        out_em[(size_t)y * DD + t] = v * inv;
        __syncthreads();
    }
}

extern "C" void kernel_launch(void* const* d_in, const int* in_sizes, int n_in,
                              void* d_out, int out_size, void* d_ws, size_t ws_size,
                              hipStream_t stream) {
    const float* inputs  = (const float*)d_in[0];   // [128,1024] f32
    const float* em      = (const float*)d_in[1];   // [32768,1024] f32
    const int*   targets = (const int*)d_in[2];     // [128] i32
    const int*   epoch   = (const int*)d_in[3];     // [1] i32

    float* out = (float*)d_out;                     // loss | ks[128] | new_em
    float* ws  = (float*)d_ws;
    float* rowsumexp = ws;
    float* cntsel    = ws + 128;
    float* sumsel    = ws + 256;
    float* lossrow   = ws + 384;
    float* out_em    = out + 1 + BB;

    init_ws_kernel<<<1, 512, 0, stream>>>(ws);
    gemm_stats_kernel<<<CC / 128, 256, 0, stream>>>(inputs, em, rowsumexp, cntsel, sumsel);
    finalize_kernel<<<BB, 32, 0, stream>>>(inputs, em, targets,
                                           rowsumexp, cntsel, sumsel, out, lossrow);
    loss_reduce_kernel<<<1, 128, 0, stream>>>(lossrow, out);
    copy_em_kernel<<<4096, 256, 0, stream>>>((const float4*)em, out_em,
                                             (CC * DD) / 4);
    ema_update_kernel<<<1, 1024, 0, stream>>>(inputs, targets, epoch, out_em);
}